// ExcitationModule_54219667144896
// MI455X (gfx1250) — compile-verified
//
#include <hip/hip_runtime.h>
#include <math.h>

// ---------------- problem constants (match reference) ----------------
#define FS_F      44100.0f
#define ENC       1024
#define NHARM     150
#define TAUMIN    110
#define TAUMAX    630          // 44100/70
#define FRAME     1260         // 2*TAUMAX
#define STRIDE    1022         // int((T-FRAME)/(T/ENC - 1))
#define NFRM      128
#define NB        4
#define TLEN      131072
#define THRESH    0.1f
#define TWO_PI_F  6.28318530717958647692f
#define CL        519          // cmdf length = TAUMAX-1-TAUMIN
#define BIGI      0x7fffffff

// LDS signal buffer: 256 zeros front pad, frame, zero tail (reads reach
// PAD + 1256 + 752 + 3 + 15 = 2282).
#define PAD       256
#define XBN       2304
#define NSTEP     375          // t0 = -240 .. 1256 step 4

typedef float v2f __attribute__((ext_vector_type(2)));
typedef float v8f __attribute__((ext_vector_type(8)));

#if __has_builtin(__builtin_amdgcn_global_load_async_to_lds_b32) && \
    __has_builtin(__builtin_amdgcn_s_wait_asynccnt)
#define HAVE_ASYNC_LDS 1
typedef __attribute__((address_space(1))) int as1_int;   // global
typedef __attribute__((address_space(3))) int as3_int;   // LDS
#else
#define HAVE_ASYNC_LDS 0
#endif

// =====================================================================
// Kernel 1: YIN pitch per frame. One 256-thread block per (batch,frame).
// Frame staged into LDS with GLOBAL_LOAD_ASYNC_TO_LDS_B32 (ASYNCcnt path)
// when available. Autocorrelation via V_WMMA_F32_16X16X4_F32 (full f32
// precision) using the Hankel-GEMM trick; waves 0..3 split the time sweep
// and merge partial tiles into LDS with ds_add_f32 atomics.
// =====================================================================
__global__ __launch_bounds__(256) void yin_kernel(const float* __restrict__ audio,
                                                  float* __restrict__ pitch_out) {
  const int tid = threadIdx.x;
  const int blk = blockIdx.x;
  const int b   = blk >> 7;
  const int f   = blk & (NFRM - 1);

  __shared__ float xb[XBN];        // padded frame
  __shared__ float corr[TAUMAX];   // autocorrelation, lags 0..629
  __shared__ float sq[FRAME + 1];  // prefix sums of squares (sqrcs)
  __shared__ float cm[CL];         // cmdf
  __shared__ float sbuf[256];      // scan scratch
  __shared__ int   minBelow, minTau;

  const float* src = audio + (size_t)b * TLEN + (size_t)f * STRIDE;
  __builtin_prefetch(src, 0, 0);   // gfx1250 global_prefetch_b8

  // ---- zero pads (disjoint from async-written frame region) ----
  for (int i = tid; i < PAD; i += 256) xb[i] = 0.0f;
  for (int i = PAD + FRAME + tid; i < XBN; i += 256) xb[i] = 0.0f;
  for (int i = tid; i < TAUMAX; i += 256) corr[i] = 0.0f;
  if (tid == 0) { minBelow = BIGI; minTau = BIGI; sq[0] = 0.0f; }

  // ---- stage frame into LDS ----
#if HAVE_ASYNC_LDS
  for (int j = tid; j < FRAME; j += 256) {
    __builtin_amdgcn_global_load_async_to_lds_b32(
        (as1_int*)(src + j), (as3_int*)(&xb[PAD + j]), 0, 0);
  }
  __builtin_amdgcn_s_wait_asynccnt(0);
#else
  for (int j = tid; j < FRAME; j += 256) xb[PAD + j] = src[j];
#endif
  __syncthreads();

  // ---- prefix sums of squares: sq[j] = sum_{t<j} x[t]^2 (block scan) ----
  float loc[5];
  float part = 0.0f;
#pragma unroll
  for (int i = 0; i < 5; ++i) {
    int j = 5 * tid + i;
    float v = (j < FRAME) ? xb[PAD + j] : 0.0f;
    part += v * v;
    loc[i] = part;                       // inclusive within thread
  }
  sbuf[tid] = part;
  __syncthreads();
  for (int off = 1; off < 256; off <<= 1) {
    float x = sbuf[tid];
    float t = (tid >= off) ? sbuf[tid - off] : 0.0f;
    __syncthreads();
    sbuf[tid] = x + t;
    __syncthreads();
  }
  float excl = sbuf[tid] - part;
#pragma unroll
  for (int i = 0; i < 5; ++i) {
    int j = 5 * tid + i;
    if (j < FRAME) sq[j + 1] = excl + loc[i];
  }
  __syncthreads();

  // ---- autocorrelation via f32 WMMA (waves 0..3, EXEC all-ones per wave) ----
  if (tid < 128) {
    const int lane = tid & 31;
    const int wv   = tid >> 5;
    const int half = lane >> 4;    // selects K pair {0,1} vs {2,3}
    const int l16  = lane & 15;    // A row m / B column n

    v8f acc0 = {}; v8f acc1 = {}; v8f acc2 = {};
    int s0 = wv * 94;
    int s1 = s0 + 94; if (s1 > NSTEP) s1 = NSTEP;
    for (int s = s0; s < s1; ++s) {
      int t0 = -240 + 4 * s;
      // A[m,k] = x[t0 + 16m + k]  (16x4, K in {2*half, 2*half+1})
      int ab = PAD + t0 + 16 * l16 + 2 * half;
      v2f A; A.x = xb[ab]; A.y = xb[ab + 1];
      // B_j[k,n] = x[t0 + tau0_j + k + n]  (4x16 Hankel)
      int bb = PAD + t0 + 2 * half + l16;
      v2f B0; B0.x = xb[bb + 240]; B0.y = xb[bb + 241];
      v2f B1; B1.x = xb[bb + 496]; B1.y = xb[bb + 497];
      v2f B2; B2.x = xb[bb + 752]; B2.y = xb[bb + 753];
      acc0 = __builtin_amdgcn_wmma_f32_16x16x4_f32(false, A, false, B0, (short)0, acc0, false, false);
      acc1 = __builtin_amdgcn_wmma_f32_16x16x4_f32(false, A, false, B1, (short)0, acc1, false, false);
      acc2 = __builtin_amdgcn_wmma_f32_16x16x4_f32(false, A, false, B2, (short)0, acc2, false, false);
    }
    // cell (m,n) of tile j holds corr at lag  tau0_j + n - 16m
#pragma unroll
    for (int v = 0; v < 8; ++v) {
      int m    = v + 8 * half;
      int base = l16 - 16 * m;
      int l0 = 240 + base;
      int l1 = 496 + base;
      int l2 = 752 + base;
      if ((unsigned)l0 < (unsigned)TAUMAX) atomicAdd(&corr[l0], acc0[v]);
      if ((unsigned)l1 < (unsigned)TAUMAX) atomicAdd(&corr[l1], acc1[v]);
      if ((unsigned)l2 < (unsigned)TAUMAX) atomicAdd(&corr[l2], acc2[v]);
    }
  }
  __syncthreads();

  // ---- cmdf (inherently serial cumsum over 629 lags) ----
  if (tid == 0) {
    float E   = sq[FRAME];
    float cum = 0.0f;
    for (int k = 1; k < TAUMAX; ++k) {
      float dk = E + (sq[FRAME - k] - sq[k]) - 2.0f * corr[k];
      cum += dk;
      float c = dk * (float)k / fmaxf(cum, 1e-5f);
      if (k >= TAUMIN + 1) cm[k - (TAUMIN + 1)] = c;
    }
  }
  __syncthreads();

  // ---- first index below threshold (argmax of boolean == min index) ----
  for (int i = tid; i < CL; i += 256)
    if (cm[i] < THRESH) atomicMin(&minBelow, i);
  __syncthreads();
  int mb = minBelow;
  int fb = (mb != BIGI && mb > 0) ? mb : TAUMAX;   // replicate reference edge case
  // ---- first index >= fb with non-negative slope ----
  for (int i = tid; i < CL; i += 256) {
    bool slope = (i == CL - 1) ? true : (cm[i + 1] - cm[i] >= 0.0f);
    if (i >= fb && slope) atomicMin(&minTau, i);
  }
  __syncthreads();
  if (tid == 0) {
    int tau = (minTau == BIGI) ? 0 : minTau;
    pitch_out[blk] = (tau > 0) ? (FS_F / (float)(tau + TAUMIN + 1)) : 0.0f;
  }
}

// =====================================================================
// Kernel 2: per-chunk theta sums. chunk c covers samples [c*1024,(c+1)*1024)
// of a batch and uses pitch[c] (repeat-by-ENC aligns exactly).
// =====================================================================
__global__ __launch_bounds__(256) void chunk_sum_kernel(const float* __restrict__ pm_g,
                                                        const float* __restrict__ pitch,
                                                        float* __restrict__ csum) {
  const int c = blockIdx.x;
  const int b = c >> 7;
  const int f = c & (NFRM - 1);
  const int tid = threadIdx.x;
  __shared__ float sb[256];
  const float* pm = pm_g + (size_t)b * TLEN + (size_t)f * ENC;
  float s = 0.0f;
  for (int j = tid; j < ENC; j += 256) s += pm[j];
  sb[tid] = s;
  __syncthreads();
  for (int off = 128; off > 0; off >>= 1) {
    if (tid < off) sb[tid] += sb[tid + off];
    __syncthreads();
  }
  if (tid == 0) csum[c] = (TWO_PI_F * pitch[c] / FS_F) * sb[0];
}

// =====================================================================
// Kernel 3: exclusive scan of the 128 chunk sums per batch (tiny).
// =====================================================================
__global__ void chunk_offset_kernel(const float* __restrict__ csum,
                                    float* __restrict__ offs) {
  int b = threadIdx.x;
  if (b < NB) {
    float acc = 0.0f;
    for (int f = 0; f < NFRM; ++f) {
      offs[b * NFRM + f] = acc;
      acc += csum[b * NFRM + f];
    }
  }
}

// =====================================================================
// Kernel 4: in-chunk inclusive phase scan + 150-harmonic sin synthesis.
// =====================================================================
__global__ __launch_bounds__(256) void synth_kernel(const float* __restrict__ pm_g,
                                                    const float* __restrict__ pitch,
                                                    const float* __restrict__ offs,
                                                    const float* __restrict__ amplitudes,
                                                    const float* __restrict__ ratio,
                                                    float* __restrict__ out) {
  const int c = blockIdx.x;
  const int b = c >> 7;
  const int f = c & (NFRM - 1);
  const int tid = threadIdx.x;
  __shared__ float sb[256];
  __shared__ float ampL[NHARM];
  for (int h = tid; h < NHARM; h += 256) ampL[h] = amplitudes[h] * ratio[0];

  const float p    = pitch[c];
  const float coef = TWO_PI_F * p / FS_F;
  const float base = offs[c];
  const float* pm  = pm_g + (size_t)b * TLEN + (size_t)f * ENC;

  float pv[4], pref[4];
  float part = 0.0f;
#pragma unroll
  for (int i = 0; i < 4; ++i) {
    float v = pm[4 * tid + i];
    pv[i] = v;
    part += coef * v;
    pref[i] = part;                       // inclusive within thread
  }
  sb[tid] = part;
  __syncthreads();                         // also covers ampL writes
  for (int off = 1; off < 256; off <<= 1) {
    float x = sb[tid];
    float t = (tid >= off) ? sb[tid - off] : 0.0f;
    __syncthreads();
    sb[tid] = x + t;
    __syncthreads();
  }
  float excl = sb[tid] - part;

  float* o = out + (size_t)b * TLEN + (size_t)f * ENC + 4 * tid;
#pragma unroll
  for (int i = 0; i < 4; ++i) {
    float phase = base + excl + pref[i];  // inclusive cumsum of theta
    float f0 = p * pv[i];
    float sig = 0.0f;
    float hf  = 1.0f;
    for (int h = 0; h < NHARM; ++h) {
      float aa = ((f0 * hf) < 22050.0f ? 1.0f : 0.0f) + 1e-7f;
      sig = fmaf(sinf(phase * hf), ampL[h] * aa, sig);
      hf += 1.0f;
    }
    o[i] = sig;
  }
}

// =====================================================================
extern "C" void kernel_launch(void* const* d_in, const int* in_sizes, int n_in,
                              void* d_out, int out_size, void* d_ws, size_t ws_size,
                              hipStream_t stream) {
  const float* audio      = (const float*)d_in[0];
  const float* pitch_mult = (const float*)d_in[1];
  const float* amplitudes = (const float*)d_in[2];
  const float* ratio      = (const float*)d_in[3];
  float* out = (float*)d_out;

  float* ws    = (float*)d_ws;
  float* pitch = ws;                // 512 floats
  float* csum  = ws + 512;          // 512 floats
  float* offs  = ws + 1024;         // 512 floats

  yin_kernel        <<<dim3(NB * NFRM), dim3(256), 0, stream>>>(audio, pitch);
  chunk_sum_kernel  <<<dim3(NB * NFRM), dim3(256), 0, stream>>>(pitch_mult, pitch, csum);
  chunk_offset_kernel<<<dim3(1),        dim3(64),  0, stream>>>(csum, offs);
  synth_kernel      <<<dim3(NB * NFRM), dim3(256), 0, stream>>>(pitch_mult, pitch, offs,
                                                                amplitudes, ratio, out);
}